// TemporalHeteroHG_55920474194543
// MI455X (gfx1250) — compile-verified
//
#include <hip/hip_runtime.h>

typedef float v2f __attribute__((ext_vector_type(2)));
typedef float v8f __attribute__((ext_vector_type(8)));

#define HIDC 64

// ---------------------------------------------------------------- zero fill
__global__ void zero_f32_kernel(float* __restrict__ p, int n) {
    int i = blockIdx.x * blockDim.x + threadIdx.x;
    if (i < n) p[i] = 0.0f;
}

// ------------------------------------------- conv0 edge scatter: 64 thr/edge
// agg[dst][c] += x[src][c] if ts[src] <= ts[dst]
__global__ void scatter_vec_kernel(const float* __restrict__ x,
                                   const float* __restrict__ ts,
                                   const int* __restrict__ src,
                                   const int* __restrict__ dst,
                                   float* __restrict__ agg, int nwork) {
    int t = blockIdx.x * blockDim.x + threadIdx.x;
    if (t >= nwork) return;
    int e = t >> 6;
    int c = t & 63;
    int s = src[e];
    int d = dst[e];
    if (ts[s] <= ts[d]) {
        atomicAdd(&agg[(size_t)d * HIDC + c], x[(size_t)s * HIDC + c]);
    }
}

// ------------------------------------- h = 0.5*agg @ W0^T + b0  (in place)
// One wave per 16-row tile; 4 accumulators cover all 64 output channels.
// Uses V_WMMA_F32_16X16X4_F32 (full fp32 matrix core path).
__global__ void __launch_bounds__(256)
gemm_wmma_kernel(float* __restrict__ aggh, const float* __restrict__ W0,
                 const float* __restrict__ b0, int N) {
    const int lane  = threadIdx.x & 31;
    const int wave  = threadIdx.x >> 5;          // wave32
    const int tile  = blockIdx.x * 8 + wave;
    const int row0  = tile * 16;
    if (row0 >= N) return;                       // uniform per wave

    const int lo    = lane & 15;                 // M (for A) / N (for B,C,D)
    const int khalf = (lane >> 4) & 1;           // lanes 16-31 hold K+2,K+3

    int r = row0 + lo;
    if (r > N - 1) r = N - 1;                    // clamp tail reads
    const float* arow = aggh + (size_t)r * HIDC;

    v8f acc0 = {}, acc1 = {}, acc2 = {}, acc3 = {};

#pragma unroll
    for (int k0 = 0; k0 < HIDC; k0 += 4) {
        const int kb = k0 + 2 * khalf;
        // A 16x4 f32: lane(lo) = row, {VGPR0,VGPR1} = K {kb, kb+1}
        v2f a = *(const v2f*)(arow + kb);
        // B 4x16 f32: lane(lo) = col n, {VGPR0,VGPR1} = K {kb, kb+1};
        // B[k][n] = W0[n][k]  (out = agg @ W0^T)
        v2f bv0 = *(const v2f*)(W0 + (size_t)(0 * 16 + lo) * HIDC + kb);
        v2f bv1 = *(const v2f*)(W0 + (size_t)(1 * 16 + lo) * HIDC + kb);
        v2f bv2 = *(const v2f*)(W0 + (size_t)(2 * 16 + lo) * HIDC + kb);
        v2f bv3 = *(const v2f*)(W0 + (size_t)(3 * 16 + lo) * HIDC + kb);
        acc0 = __builtin_amdgcn_wmma_f32_16x16x4_f32(false, a, false, bv0,
                                                     (short)0, acc0, false, false);
        acc1 = __builtin_amdgcn_wmma_f32_16x16x4_f32(false, a, false, bv1,
                                                     (short)0, acc1, false, false);
        acc2 = __builtin_amdgcn_wmma_f32_16x16x4_f32(false, a, false, bv2,
                                                     (short)0, acc2, false, false);
        acc3 = __builtin_amdgcn_wmma_f32_16x16x4_f32(false, a, false, bv3,
                                                     (short)0, acc3, false, false);
    }

    // C/D layout: lanes 0-15 -> rows row0+v, lanes 16-31 -> rows row0+8+v
    const int rbase = row0 + 8 * khalf;
    const float bb0 = b0[0 * 16 + lo];
    const float bb1 = b0[1 * 16 + lo];
    const float bb2 = b0[2 * 16 + lo];
    const float bb3 = b0[3 * 16 + lo];
#pragma unroll
    for (int v = 0; v < 8; ++v) {
        const int rr = rbase + v;
        if (rr < N) {
            float* orow = aggh + (size_t)rr * HIDC;
            orow[0 * 16 + lo] = acc0[v] * 0.5f + bb0;
            orow[1 * 16 + lo] = acc1[v] * 0.5f + bb1;
            orow[2 * 16 + lo] = acc2[v] * 0.5f + bb2;
            orow[3 * 16 + lo] = acc3[v] * 0.5f + bb3;
        }
    }
}

// ---------------------------------------- s[n] = dot(h[n], W1)  (wave/node)
__global__ void dot_w1_kernel(const float* __restrict__ h,
                              const float* __restrict__ W1,
                              float* __restrict__ s, int N) {
    const int lane = threadIdx.x & 31;
    const int node = blockIdx.x * 8 + (threadIdx.x >> 5);
    if (node >= N) return;                       // uniform per wave
    const float* hr = h + (size_t)node * HIDC;
    float p = hr[lane] * W1[lane] + hr[lane + 32] * W1[lane + 32];
#pragma unroll
    for (int m = 16; m >= 1; m >>= 1) p += __shfl_xor(p, m, 32);
    if (lane == 0) s[node] = p;
}

// ------------------------------------------- conv1 edge scatter (scalar)
__global__ void scatter_scalar_kernel(const float* __restrict__ s,
                                      const float* __restrict__ ts,
                                      const int* __restrict__ src,
                                      const int* __restrict__ dst,
                                      float* __restrict__ agg1, int E) {
    int e = blockIdx.x * blockDim.x + threadIdx.x;
    if (e >= E) return;
    int a = src[e];
    int d = dst[e];
    if (ts[a] <= ts[d]) atomicAdd(&agg1[d], s[a]);
}

// --------------------------------------------------- fused FC head (rank-1)
__global__ void head_kernel(const float* __restrict__ agg1,
                            const float* __restrict__ b1,
                            const float* __restrict__ fc1W,
                            const float* __restrict__ fc1b,
                            const float* __restrict__ fc2W,
                            const float* __restrict__ fc2b,
                            float* __restrict__ out, int N) {
    int n = blockIdx.x * blockDim.x + threadIdx.x;
    if (n >= N) return;
    const float h1 = agg1[n] * 0.5f + b1[0];
    float o0 = fc2b[0];
    float o1 = fc2b[1];
#pragma unroll 8
    for (int k = 0; k < HIDC; ++k) {
        float t = h1 * fc1W[k] + fc1b[k];
        t = t > 0.0f ? t : 0.0f;
        o0 += t * fc2W[k];
        o1 += t * fc2W[HIDC + k];
    }
    out[2 * n + 0] = o0;
    out[2 * n + 1] = o1;
}

extern "C" void kernel_launch(void* const* d_in, const int* in_sizes, int n_in,
                              void* d_out, int out_size, void* d_ws, size_t ws_size,
                              hipStream_t stream) {
    const float* x    = (const float*)d_in[0];
    const float* ts   = (const float*)d_in[1];
    const int*   eia  = (const int*)d_in[2];
    const int*   eib  = (const int*)d_in[3];
    const float* W0   = (const float*)d_in[4];
    const float* b0   = (const float*)d_in[5];
    const float* W1   = (const float*)d_in[6];
    const float* b1   = (const float*)d_in[7];
    const float* fc1W = (const float*)d_in[8];
    const float* fc1b = (const float*)d_in[9];
    const float* fc2W = (const float*)d_in[10];
    const float* fc2b = (const float*)d_in[11];

    const int N = in_sizes[1];          // 100000
    const int E = in_sizes[2] / 2;      // 1600000

    const int* srcA = eia;
    const int* dstA = eia + E;
    const int* srcB = eib;
    const int* dstB = eib + E;

    // workspace: agg0/h [N*64] | agg1 [N] | s [N]   (~26.5 MB)
    float* aggh = (float*)d_ws;
    float* agg1 = aggh + (size_t)N * HIDC;
    float* svec = agg1 + N;

    // zero agg0 and agg1 (contiguous)
    const int nzero = N * HIDC + N;
    zero_f32_kernel<<<(nzero + 255) / 256, 256, 0, stream>>>(aggh, nzero);

    // conv0 scatter (both edge types into one accumulator)
    const int nwork = E * HIDC;
    scatter_vec_kernel<<<(nwork + 255) / 256, 256, 0, stream>>>(x, ts, srcA, dstA, aggh, nwork);
    scatter_vec_kernel<<<(nwork + 255) / 256, 256, 0, stream>>>(x, ts, srcB, dstB, aggh, nwork);

    // h = 0.5*agg @ W0^T + b0 via fp32 WMMA (in place)
    gemm_wmma_kernel<<<(N + 127) / 128, 256, 0, stream>>>(aggh, W0, b0, N);

    // per-node scalar s = h . W1
    dot_w1_kernel<<<(N + 7) / 8, 256, 0, stream>>>(aggh, W1, svec, N);

    // conv1 scatter (scalar atomics)
    scatter_scalar_kernel<<<(E + 255) / 256, 256, 0, stream>>>(svec, ts, srcA, dstA, agg1, E);
    scatter_scalar_kernel<<<(E + 255) / 256, 256, 0, stream>>>(svec, ts, srcB, dstB, agg1, E);

    // fused FC head
    head_kernel<<<(N + 255) / 256, 256, 0, stream>>>(agg1, b1, fc1W, fc1b, fc2W, fc2b,
                                                     (float*)d_out, N);
}